// BlockLearnableCompressionMatrix_51170240365301
// MI455X (gfx1250) — compile-verified
//
#include <hip/hip_runtime.h>

// ---------------------------------------------------------------------------
// BlockLearnableCompressionMatrix for MI455X (gfx1250, wave32).
//
// Outputs (concatenated in d_out, fp32):
//   Ax     : (16, 128, 2048)  = per-block weighted channel reduction
//   A_full : (128, 1024)      = block-diagonal expansion of W
//
// Ax[b,m,t] = sum_{c<8} W[m,c] * x[b, 8m+c, t]
//
// HBM-bound (~151 MB @ 23.3 TB/s -> ~6.5 us floor; 67 MFLOP is negligible).
// The per-block dots ride the matrix pipe via V_WMMA_F32_16X16X4_F32 with a
// broadcast-row A tile (every D row identical), so DRAM traffic stays minimal.
// Two t-tiles per iteration: lanes 0-15 take tile A's row M=0 (acc VGPR0),
// lanes 16-31 take tile B's row M=8 (also acc VGPR0) -> one unpredicated,
// fully coalesced 128B store per iteration, zero EXEC manipulation.
// ---------------------------------------------------------------------------

typedef __attribute__((ext_vector_type(2))) float v2f;   // A/B tile: 2 VGPRs
typedef __attribute__((ext_vector_type(8))) float v8f;   // C/D tile: 8 VGPRs

#define C_BLOCK 8
#define N_CH    1024
#define M_OUT   128      // N_CH / C_BLOCK
#define T_LEN   2048
#define B_SZ    16

// One wave32 per (b, m, t-strip). Each strip = 8 iterations x 32 t = 256 t.
// 16*128*8 = 16384 waves = 2048 blocks x 8 waves.
__global__ __launch_bounds__(256)
void blkcomp_wmma_kernel(const float* __restrict__ x,
                         const float* __restrict__ W,
                         float* __restrict__ Ax)
{
    const int lane  = threadIdx.x & 31;
    const int w     = blockIdx.x * 8 + (threadIdx.x >> 5);
    const int strip = w & 7;                 // t-strip within (b, m)
    const int bm    = w >> 3;
    const int mi    = bm & (M_OUT - 1);
    const int b     = bm >> 7;
    const int tbase = strip * 256;

    const int h = lane >> 4;                 // half-wave selects K pair
    const int n = lane & 15;                 // N (t offset within tile)

    // A-matrix (16x4 f32), broadcast: every row i carries W[mi, cbase+k].
    // Layout: lane l, VGPR v  ->  A[M = l%16, K = 2*(l/16) + v].
    const float* wrow = W + mi * C_BLOCK;
    v2f a0, a1;
    a0.x = wrow[2 * h + 0];                  // c = 0..3
    a0.y = wrow[2 * h + 1];
    a1.x = wrow[4 + 2 * h + 0];              // c = 4..7
    a1.y = wrow[4 + 2 * h + 1];

    const long xbase = ((long)b * N_CH + (long)mi * C_BLOCK) * T_LEN;
    const long obase = ((long)b * M_OUT + mi) * T_LEN;

    for (int j = 0; j < 8; ++j) {
        const int t0 = tbase + 32 * j;       // two 16-t tiles: [t0, t0+16)
        const float* xc = x + xbase + t0 + n;

        // B tiles (4x16 f32): lane l, VGPR v -> B[K = 2*(l/16)+v, N = l%16]
        //                     = x[b, 8*mi + cbase + K, t0(+16) + N]
        v2f b0a, b1a, b0b, b1b;
        b0a.x = xc[(2 * h + 0) * T_LEN];
        b0a.y = xc[(2 * h + 1) * T_LEN];
        b1a.x = xc[(4 + 2 * h + 0) * T_LEN];
        b1a.y = xc[(4 + 2 * h + 1) * T_LEN];
        b0b.x = xc[(2 * h + 0) * T_LEN + 16];
        b0b.y = xc[(2 * h + 1) * T_LEN + 16];
        b1b.x = xc[(4 + 2 * h + 0) * T_LEN + 16];
        b1b.y = xc[(4 + 2 * h + 1) * T_LEN + 16];

        // Prefetch next iteration's lines (global_prefetch_b8).
        if (j < 7) {
            __builtin_prefetch(xc + 32, 0, 0);
            __builtin_prefetch(xc + 32 + 4 * T_LEN, 0, 0);
        }

        // Broadcast-row A => all 16 rows of each D are identical.
        v8f accA = {}, accB = {};
        accA = __builtin_amdgcn_wmma_f32_16x16x4_f32(
                   false, a0, false, b0a, (short)0, accA, false, false);
        accA = __builtin_amdgcn_wmma_f32_16x16x4_f32(
                   false, a1, false, b1a, (short)0, accA, false, false);
        accB = __builtin_amdgcn_wmma_f32_16x16x4_f32(
                   false, a0, false, b0b, (short)0, accB, false, false);
        accB = __builtin_amdgcn_wmma_f32_16x16x4_f32(
                   false, a1, false, b1b, (short)0, accB, false, false);

        // acc VGPR0: lanes 0-15 = D[M=0, N=lane] (tile A),
        //            lanes 16-31 = D[M=8, N=lane-16] (tile B, rows identical).
        // => lane covers t = t0 + lane; one full-wave contiguous 128B store.
        const float outv = (lane < 16) ? accA[0] : accB[0];
        Ax[obase + t0 + lane] = outv;
    }
}

// A_full[mm, ch] = (ch/8 == mm) ? W[mm, ch%8] : 0   — 0.5 MB, trivial.
__global__ __launch_bounds__(256)
void build_A_kernel(const float* __restrict__ W, float* __restrict__ Afull)
{
    int idx = blockIdx.x * blockDim.x + threadIdx.x;
    if (idx >= M_OUT * N_CH) return;
    int mm = idx >> 10;          // / N_CH
    int ch = idx & (N_CH - 1);
    float v = 0.0f;
    if ((ch >> 3) == mm) v = W[mm * C_BLOCK + (ch & 7)];
    Afull[idx] = v;
}

extern "C" void kernel_launch(void* const* d_in, const int* in_sizes, int n_in,
                              void* d_out, int out_size, void* d_ws, size_t ws_size,
                              hipStream_t stream)
{
    const float* x = (const float*)d_in[0];   // (16, 1024, 2048) fp32
    const float* W = (const float*)d_in[1];   // (128, 8) fp32

    float* Ax    = (float*)d_out;                                   // 4194304
    float* Afull = Ax + (size_t)B_SZ * M_OUT * T_LEN;               //  131072

    blkcomp_wmma_kernel<<<dim3(2048), dim3(256), 0, stream>>>(x, W, Ax);

    const int nA = M_OUT * N_CH;
    build_A_kernel<<<dim3((nA + 255) / 256), dim3(256), 0, stream>>>(W, Afull);
}